// SkeletonMergerHead_81750407512431
// MI455X (gfx1250) — compile-verified
//
#include <hip/hip_runtime.h>
#include <hip/hip_bf16.h>
#include <math.h>

// ---------------- problem constants (match reference) ----------------
#define BATCH 32
#define NPTS  16384
#define KDIM  16
#define NPAIR 120   // K*(K-1)/2

typedef __attribute__((ext_vector_type(2))) float v2f;
typedef __attribute__((ext_vector_type(4))) float v4f;
typedef __attribute__((ext_vector_type(8))) float v8f;

#if defined(__HIP_DEVICE_COMPILE__) && __has_builtin(__builtin_amdgcn_wmma_f32_16x16x4_f32)
#define HAVE_WMMA_F32X4 1
#else
#define HAVE_WMMA_F32X4 0
#endif

// One K=4 step of D(16x16) += A(16x4) * B(4x16), all f32, wave32.
__device__ inline v8f wmma_step(v2f a, v2f w, v8f c) {
#if HAVE_WMMA_F32X4
  return __builtin_amdgcn_wmma_f32_16x16x4_f32(false, a, false, w, (short)0, c,
                                               false, false);
#else
  // host-pass placeholder (never used for device results)
  c[0] += a.x * w.x + a.y * w.y;
  return c;
#endif
}

// B-fragments of ptl_w (16x16) for the 4 chained K=4 steps.
// B(4x16) layout: VGPR v, lane L: col j = L&15, row K = 4*s + (L>>4)*2 + v
__device__ inline void load_wfrags(const float* __restrict__ W, int j, int kh,
                                   v2f wf[4]) {
#pragma unroll
  for (int s = 0; s < 4; ++s) {
    int r = 4 * s + kh * 2;
    wf[s].x = W[r * KDIM + j];
    wf[s].y = W[(r + 1) * KDIM + j];
  }
}

// A(16x4) layout: lanes 0-15 -> M = lane, K = {0,1}; lanes 16-31 -> K = {2,3}.
// row points at KP[b, n0 + (lane&15), kh*2]; step s reads k = 4s + kh*2 (+1).
__device__ inline v8f kpl_tile(const float* __restrict__ row, const v2f wf[4]) {
  v8f d = {};
#pragma unroll
  for (int s = 0; s < 4; ++s) {
    v2f a = *(const v2f*)(row + 4 * s);  // 8B aligned (kh*2 + 4s even)
    d = wmma_step(a, wf[s], d);
  }
  return d;
}

// ---- K1: online-softmax statistics pass --------------------------------
// grid = BATCH*32 blocks of 256 (8 waves); each wave does 4 16-row tiles.
// Produces per-block (m, s, sx, sy, sz) per column k: running max + rescaled
// sums of e = exp(KPL - m) and e * x_{x,y,z}. One read of KP + x, no KPL spill.
__global__ __launch_bounds__(256) void stats_kernel(
    const float* __restrict__ KP, const float* __restrict__ W,
    const float* __restrict__ X, float* __restrict__ pstat /*[1024][16][5]*/) {
  const int lane = threadIdx.x & 31;
  const int wave = threadIdx.x >> 5;
  const int blk = blockIdx.x & 31;
  const int b = blockIdx.x >> 5;
  const int j = lane & 15;   // output column k (fixed per lane in C/D layout)
  const int kh = lane >> 4;  // half-wave selector

  v2f wf[4];
  load_wfrags(W, j, kh, wf);

  float m = -3.402823466e38f, s = 0.f, sx = 0.f, sy = 0.f, sz = 0.f;
#pragma unroll
  for (int q = 0; q < 4; ++q) {
    const int n0 = blk * 512 + q * 128 + wave * 16;
    const float* row = KP + ((size_t)(b * NPTS + n0 + j)) * KDIM + kh * 2;
    v8f d = kpl_tile(row, wf);

    const int rbase = n0 + kh * 8;  // first row held by this lane
    // x rows rbase..rbase+7: 24 contiguous floats, 16B aligned
    const float* xp = X + ((size_t)(b * NPTS) + rbase) * 3;
    float xs[24];
#pragma unroll
    for (int u = 0; u < 6; ++u) {
      v4f t4 = *(const v4f*)(xp + 4 * u);
      xs[4 * u + 0] = t4.x;
      xs[4 * u + 1] = t4.y;
      xs[4 * u + 2] = t4.z;
      xs[4 * u + 3] = t4.w;
    }

    float tm = d[0];
#pragma unroll
    for (int v = 1; v < 8; ++v) tm = fmaxf(tm, d[v]);
    float mnew = fmaxf(m, tm);
    float scale = expf(m - mnew);  // exp(-huge) -> 0 on first tile
    s *= scale; sx *= scale; sy *= scale; sz *= scale;
#pragma unroll
    for (int v = 0; v < 8; ++v) {
      float e = expf(d[v] - mnew);
      s += e;
      sx += e * xs[3 * v + 0];
      sy += e * xs[3 * v + 1];
      sz += e * xs[3 * v + 2];
    }
    m = mnew;
  }

  // fold the two half-waves (same column j, other 8-row group)
  float mo = __shfl_xor(m, 16, 32);
  float so = __shfl_xor(s, 16, 32);
  float sxo = __shfl_xor(sx, 16, 32);
  float syo = __shfl_xor(sy, 16, 32);
  float szo = __shfl_xor(sz, 16, 32);
  float M = fmaxf(m, mo);
  float ca = expf(m - M), cb = expf(mo - M);
  s = s * ca + so * cb;
  sx = sx * ca + sxo * cb;
  sy = sy * ca + syo * cb;
  sz = sz * ca + szo * cb;
  m = M;

  __shared__ float st[8][16][5];
  if (lane < 16) {
    st[wave][j][0] = m;
    st[wave][j][1] = s;
    st[wave][j][2] = sx;
    st[wave][j][3] = sy;
    st[wave][j][4] = sz;
  }
  __syncthreads();
  if (threadIdx.x < 16) {
    int j2 = threadIdx.x;
    float Mb = st[0][j2][0], Sb = st[0][j2][1];
    float nx = st[0][j2][2], ny = st[0][j2][3], nz = st[0][j2][4];
#pragma unroll
    for (int w2 = 1; w2 < 8; ++w2) {
      float m2 = st[w2][j2][0];
      float Mn = fmaxf(Mb, m2);
      float a = expf(Mb - Mn), c = expf(m2 - Mn);
      Sb = Sb * a + st[w2][j2][1] * c;
      nx = nx * a + st[w2][j2][2] * c;
      ny = ny * a + st[w2][j2][3] * c;
      nz = nz * a + st[w2][j2][4] * c;
      Mb = Mn;
    }
    float* p = pstat + ((size_t)blockIdx.x * 16 + j2) * 5;
    p[0] = Mb; p[1] = Sb; p[2] = nx; p[3] = ny; p[4] = nz;
  }
}

// ---- K2: merge block partials -> M, 1/S, KPCD; counts; offsets ----------
__global__ __launch_bounds__(512) void finalize_kernel(
    const float* __restrict__ pstat, float* __restrict__ MX,
    float* __restrict__ SINV, float* __restrict__ kpcd_ws,
    float* __restrict__ kpcd_out, int* __restrict__ counts,
    int* __restrict__ offs) {
  __shared__ float kp[BATCH * KDIM * 3];
  int idx = threadIdx.x;  // 512 = BATCH*KDIM
  if (idx < BATCH * KDIM) {
    int b = idx >> 4, j = idx & 15;
    float M = -3.402823466e38f, S = 0.f, nx = 0.f, ny = 0.f, nz = 0.f;
    for (int t = 0; t < 32; ++t) {
      const float* p = pstat + ((size_t)((b * 32 + t) * 16) + j) * 5;
      float m2 = p[0];
      float Mn = fmaxf(M, m2);
      float a = expf(M - Mn), c = expf(m2 - Mn);
      S = S * a + p[1] * c;
      nx = nx * a + p[2] * c;
      ny = ny * a + p[3] * c;
      nz = nz * a + p[4] * c;
      M = Mn;
    }
    float inv = 1.0f / S;
    MX[idx] = M;
    SINV[idx] = inv;
    float cx = nx * inv, cy = ny * inv, cz = nz * inv;
    kp[idx * 3 + 0] = cx; kp[idx * 3 + 1] = cy; kp[idx * 3 + 2] = cz;
    kpcd_ws[idx * 3 + 0] = cx; kpcd_ws[idx * 3 + 1] = cy; kpcd_ws[idx * 3 + 2] = cz;
    kpcd_out[idx * 3 + 0] = cx; kpcd_out[idx * 3 + 1] = cy; kpcd_out[idx * 3 + 2] = cz;
  }
  __syncthreads();
  if (idx < NPAIR) {
    int p = idx;
    int pi = 1, pj = 0, c = 0;
    for (int i = 0; i < KDIM; ++i)
      for (int j2 = 0; j2 < i; ++j2) {
        if (c == p) { pi = i; pj = j2; }
        ++c;
      }
    float acc = 0.f;
    for (int b = 0; b < BATCH; ++b) {
      const float* a = kp + (b * KDIM + pi) * 3;
      const float* q = kp + (b * KDIM + pj) * 3;
      float dx = a[0] - q[0], dy = a[1] - q[1], dz = a[2] - q[2];
      acc += sqrtf(0.001f + dx * dx + dy * dy + dz * dz);
    }
    float dist = acc / (float)BATCH;
    int cnt = (int)((double)dist / 0.01);  // truncation, matches astype(int32)
    cnt = cnt < 15 ? 15 : (cnt > 200 ? 200 : cnt);
    counts[p] = cnt;
  }
  __syncthreads();
  if (idx == 0) {
    int o = 0;
    for (int p = 0; p < NPAIR; ++p) { offs[p] = o; o += counts[p]; }
    offs[NPAIR] = o;
  }
}

// ---- K3: recompute tiles, write normalized KPA directly -----------------
// KPA[b, k, n] = exp(KPL - M[b,k]) * (1/S[b,k]); contiguous 8-float stores.
__global__ __launch_bounds__(256) void kpa_write_kernel(
    const float* __restrict__ KP, const float* __restrict__ W,
    const float* __restrict__ MX, const float* __restrict__ SINV,
    float* __restrict__ kpa) {
  const int lane = threadIdx.x & 31;
  const int wave = threadIdx.x >> 5;
  const int blk = blockIdx.x & 31;
  const int b = blockIdx.x >> 5;
  const int j = lane & 15;
  const int kh = lane >> 4;

  v2f wf[4];
  load_wfrags(W, j, kh, wf);
  const float mj = MX[b * KDIM + j];
  const float inv = SINV[b * KDIM + j];

#pragma unroll
  for (int q = 0; q < 4; ++q) {
    const int n0 = blk * 512 + q * 128 + wave * 16;
    const float* row = KP + ((size_t)(b * NPTS + n0 + j)) * KDIM + kh * 2;
    v8f d = kpl_tile(row, wf);

    const int rbase = n0 + kh * 8;
    float ev[8];
#pragma unroll
    for (int v = 0; v < 8; ++v) ev[v] = expf(d[v] - mj) * inv;
    float* op = kpa + (size_t)b * KDIM * NPTS + (size_t)j * NPTS + rbase;
    v4f o0 = {ev[0], ev[1], ev[2], ev[3]};
    v4f o1 = {ev[4], ev[5], ev[6], ev[7]};
    *(v4f*)(op) = o0;
    *(v4f*)(op + 4) = o1;
  }
}

// ---- K4: RP_concat[b, offs[p]+t, :] = emb[p,t,:] + A*f + B*(1-f) --------
__global__ void rp_kernel(const float* __restrict__ kpcd,
                          const float* __restrict__ emb,
                          const int* __restrict__ counts,
                          const int* __restrict__ offs, float* __restrict__ rp,
                          long long total) {
  int p = blockIdx.x, b = blockIdx.y, t = threadIdx.x;
  int c = counts[p];
  if (t >= c) return;
  int pi = 1, pj = 0, cc = 0;
  for (int i = 0; i < KDIM; ++i)
    for (int j2 = 0; j2 < i; ++j2) {
      if (cc == p) { pi = i; pj = j2; }
      ++cc;
    }
  float f = (c > 1) ? (float)t / (float)(c - 1) : 0.0f;
  const float* A = kpcd + (b * KDIM + pi) * 3;
  const float* Bq = kpcd + (b * KDIM + pj) * 3;
  float* o = rp + ((size_t)b * (size_t)total + (size_t)(offs[p] + t)) * 3;
  const float* e = emb + ((size_t)p * 200 + t) * 3;
  o[0] = e[0] + A[0] * f + Bq[0] * (1.0f - f);
  o[1] = e[1] + A[1] * f + Bq[1] * (1.0f - f);
  o[2] = e[2] + A[2] * f + Bq[2] * (1.0f - f);
}

// ---------------- MLP tail (tiny) ----------------
__global__ void gfp_kernel(const float* __restrict__ GF,
                           float* __restrict__ gfp) {
  int idx = blockIdx.x * blockDim.x + threadIdx.x;
  if (idx >= BATCH * 1024) return;
  const float* g = GF + (size_t)idx * 16;
  float m = g[0];
#pragma unroll
  for (int s = 1; s < 16; ++s) m = fmaxf(m, g[s]);
  gfp[idx] = m;
}

__global__ void fc_kernel(const float* __restrict__ in,
                          const float* __restrict__ w,
                          const float* __restrict__ bias,
                          float* __restrict__ out, int in_dim, int out_dim) {
  int idx = blockIdx.x * blockDim.x + threadIdx.x;
  if (idx >= BATCH * out_dim) return;
  int b = idx / out_dim, o = idx % out_dim;
  const float* x = in + (size_t)b * in_dim;
  float acc = bias[o];
  for (int i = 0; i < in_dim; ++i) acc += x[i] * w[(size_t)i * out_dim + o];
  out[idx] = acc;
}

__global__ void bn_relu_kernel(const float* __restrict__ z,
                               const float* __restrict__ g,
                               const float* __restrict__ beta,
                               float* __restrict__ h, int dim) {
  int o = blockIdx.x * blockDim.x + threadIdx.x;
  if (o >= dim) return;
  float mu = 0.f;
  for (int b = 0; b < BATCH; ++b) mu += z[(size_t)b * dim + o];
  mu /= (float)BATCH;
  float var = 0.f;
  for (int b = 0; b < BATCH; ++b) {
    float d = z[(size_t)b * dim + o] - mu;
    var += d * d;
  }
  var /= (float)BATCH;
  float inv = rsqrtf(var + 1e-5f);
  float gg = g[o], bb = beta[o];
  for (int b = 0; b < BATCH; ++b) {
    float v = gg * (z[(size_t)b * dim + o] - mu) * inv + bb;
    h[(size_t)b * dim + o] = v > 0.f ? v : 0.f;
  }
}

__global__ void ma_kernel(const float* __restrict__ h,
                          const float* __restrict__ w,
                          const float* __restrict__ bias,
                          float* __restrict__ out) {
  int idx = blockIdx.x * blockDim.x + threadIdx.x;
  if (idx >= BATCH * NPAIR) return;
  int b = idx / NPAIR, o = idx % NPAIR;
  const float* x = h + (size_t)b * 256;
  float acc = bias[o];
  for (int i = 0; i < 256; ++i) acc += x[i] * w[i * NPAIR + o];
  out[idx] = 1.0f / (1.0f + expf(-acc));
}

__global__ void lf_kernel(const float* __restrict__ emb,
                          float* __restrict__ lf) {
  int idx = blockIdx.x * blockDim.x + threadIdx.x;
  if (idx >= 200 * NPAIR * 3) return;
  int c = idx / (NPAIR * 3);
  int r = idx % (NPAIR * 3);
  int p = r / 3, d = r % 3;
  lf[idx] = emb[((size_t)p * 200 + c) * 3 + d];
}

// ---------------- host launcher ----------------
extern "C" void kernel_launch(void* const* d_in, const int* in_sizes, int n_in,
                              void* d_out, int out_size, void* d_ws,
                              size_t ws_size, hipStream_t stream) {
  (void)in_sizes; (void)n_in; (void)ws_size;
  const float* input_x = (const float*)d_in[0];
  const float* KP = (const float*)d_in[1];
  const float* GF = (const float*)d_in[2];
  const float* Wp = (const float*)d_in[3];
  // d_in[4] = ptl_b: constant along the softmax axis -> cancels; unused.
  const float* w1 = (const float*)d_in[5];
  const float* b1 = (const float*)d_in[6];
  const float* g1 = (const float*)d_in[7];
  const float* bb1 = (const float*)d_in[8];
  const float* w2 = (const float*)d_in[9];
  const float* b2 = (const float*)d_in[10];
  const float* g2 = (const float*)d_in[11];
  const float* bb2 = (const float*)d_in[12];
  const float* wl = (const float*)d_in[13];
  const float* bl = (const float*)d_in[14];
  const float* emb = (const float*)d_in[15];

  // Recover total = sum(counts) from out_size; outputs are flat-concatenated:
  // RP (B*total*3) | KPCD (B*16*3) | KPA (B*16*N) | LF (200*360) | MA (B*120)
  const long long fixedElems = (long long)BATCH * KDIM * 3 +
                               (long long)BATCH * KDIM * NPTS +
                               200LL * NPAIR * 3 + (long long)BATCH * NPAIR;
  const long long total =
      ((long long)out_size - fixedElems) / ((long long)BATCH * 3);

  float* out = (float*)d_out;
  float* rp = out;
  float* kpcd = out + (size_t)BATCH * (size_t)total * 3;
  float* kpa = kpcd + BATCH * KDIM * 3;
  float* lf = kpa + (size_t)BATCH * KDIM * NPTS;
  float* ma = lf + 200 * NPAIR * 3;

  // workspace carve-up (floats), ~667 KB
  float* wsf = (float*)d_ws;
  float* PSTAT = wsf;                 // 1024*16*5 = 81920
  float* MXA = PSTAT + 81920;         // 512
  float* SINV = MXA + 512;            // 512
  float* KPCDW = SINV + 512;          // 1536
  int* CNT = (int*)(KPCDW + 1536);    // 128
  int* OFF = CNT + 128;               // 128
  float* GFPW = (float*)(OFF + 128);  // 32768
  float* Z1 = GFPW + 32768;           // 16384
  float* H1 = Z1 + 16384;             // 16384
  float* Z2 = H1 + 16384;             // 8192
  float* H2 = Z2 + 8192;              // 8192

  stats_kernel<<<BATCH * 32, 256, 0, stream>>>(KP, Wp, input_x, PSTAT);
  finalize_kernel<<<1, 512, 0, stream>>>(PSTAT, MXA, SINV, KPCDW, kpcd, CNT, OFF);
  kpa_write_kernel<<<BATCH * 32, 256, 0, stream>>>(KP, Wp, MXA, SINV, kpa);
  {
    dim3 g(NPAIR, BATCH);
    rp_kernel<<<g, 256, 0, stream>>>(KPCDW, emb, CNT, OFF, rp, total);
  }
  gfp_kernel<<<(BATCH * 1024 + 255) / 256, 256, 0, stream>>>(GF, GFPW);
  fc_kernel<<<(BATCH * 512 + 255) / 256, 256, 0, stream>>>(GFPW, w1, b1, Z1, 1024, 512);
  bn_relu_kernel<<<2, 256, 0, stream>>>(Z1, g1, bb1, H1, 512);
  fc_kernel<<<(BATCH * 256 + 255) / 256, 256, 0, stream>>>(H1, w2, b2, Z2, 512, 256);
  bn_relu_kernel<<<1, 256, 0, stream>>>(Z2, g2, bb2, H2, 256);
  ma_kernel<<<(BATCH * NPAIR + 255) / 256, 256, 0, stream>>>(H2, wl, bl, ma);
  lf_kernel<<<(200 * NPAIR * 3 + 255) / 256, 256, 0, stream>>>(emb, lf);
}